// GAT_35476429865193
// MI455X (gfx1250) — compile-verified
//
#include <hip/hip_runtime.h>

#define N_NODES 50000
#define N_EDGES 800000
#define ETOT    (N_EDGES + N_NODES)   /* 850000, with self-loops */
#define IN_F    128
#define HID     32
#define HEADS   8
#define F1      (HEADS * HID)         /* 256 */
#define CLASSES 2
#define NEG_SLOPE 0.2f

typedef __attribute__((ext_vector_type(2))) float v2f;
typedef __attribute__((ext_vector_type(8))) float v8f;

// ---------- helpers ----------
__device__ __forceinline__ float leaky(float x) { return x >= 0.f ? x : NEG_SLOPE * x; }

// Order-preserving float<->int key (involution): signed-int compare == float compare.
__device__ __forceinline__ int f2ord(float f) {
  int i = __float_as_int(f);
  return (i < 0) ? (i ^ 0x7FFFFFFF) : i;
}
__device__ __forceinline__ float ord2f(int k) {
  return __int_as_float((k < 0) ? (k ^ 0x7FFFFFFF) : k);
}

__device__ __forceinline__ void edge_sd(const long long* __restrict__ ei, int e, int& s, int& d) {
  if (e < N_EDGES) { s = (int)ei[e]; d = (int)ei[N_EDGES + e]; }
  else             { s = d = e - N_EDGES; }          // appended self-loops
}

// ---------- init: zero accumulators, -inf maxes, seed out with bias2 ----------
__global__ void k_init(float* __restrict__ agg1, float* __restrict__ s1, int* __restrict__ m1,
                       float* __restrict__ s2, int* __restrict__ m2,
                       float* __restrict__ out, const float* __restrict__ b2) {
  long long i = (long long)blockIdx.x * blockDim.x + threadIdx.x;
  if (i < (long long)N_NODES * F1) agg1[i] = 0.f;
  if (i < N_NODES * HEADS) { s1[i] = 0.f; m1[i] = (int)0x80000000; }
  if (i < N_NODES)         { s2[i] = 0.f; m2[i] = (int)0x80000000; }
  if (i < N_NODES * CLASSES) out[i] = b2[i & 1];
}

// ---------- layer-1 GEMM: h1 = x @ W1 via V_WMMA_F32_16X16X4_F32 ----------
// One wave -> one 16x16 tile of h1.  50000/16=3125 M-tiles, 256/16=16 N-tiles.
__global__ void k_gemm1(const float* __restrict__ x, const float* __restrict__ W1,
                        float* __restrict__ h1) {
  const int lane  = threadIdx.x & 31;
  const int wave  = threadIdx.x >> 5;
  const int tile  = blockIdx.x * (blockDim.x >> 5) + wave;
  const int NT    = F1 / 16;                       // 16
  const int mt    = tile / NT;
  const int nt    = tile % NT;
  if (mt >= N_NODES / 16) return;                  // uniform per wave

  // A: 16x4 f32 — lane&15 = M row, (lane>>4)*2 = K pair base
  const int m   = mt * 16 + (lane & 15);
  const int bk  = (lane >> 4) * 2;
  // B: 4x16 f32 — lane&15 = N col, same half-wave K split as A
  const int n   = nt * 16 + (lane & 15);

  const float* __restrict__ xr = x + (size_t)m * IN_F;
  v8f c = {};
  #pragma unroll 4
  for (int k = 0; k < IN_F; k += 4) {
    v2f a = *(const v2f*)(xr + k + bk);            // x[m][k+bk], x[m][k+bk+1]
    v2f b;
    b.x = W1[(size_t)(k + bk)     * F1 + n];
    b.y = W1[(size_t)(k + bk + 1) * F1 + n];
    c = __builtin_amdgcn_wmma_f32_16x16x4_f32(false, a, false, b, (short)0, c, false, false);
  }
  // C/D: VGPR v -> rows v (lanes 0-15) / v+8 (lanes 16-31), N = lane&15
  const int mBase = mt * 16 + 8 * (lane >> 4);
  const int nOut  = nt * 16 + (lane & 15);
  #pragma unroll
  for (int v = 0; v < 8; ++v)
    h1[(size_t)(mBase + v) * F1 + nOut] = c[v];
}

// ---------- per-(node,head) attention logits ----------
__global__ void k_att1(const float* __restrict__ h1, const float* __restrict__ att_s,
                       const float* __restrict__ att_d,
                       float* __restrict__ a_s, float* __restrict__ a_d) {
  int idx = blockIdx.x * blockDim.x + threadIdx.x;   // n*8 + h
  if (idx >= N_NODES * HEADS) return;
  const int h = idx & 7, n = idx >> 3;
  const float* __restrict__ row = h1 + (size_t)n * F1 + h * HID;
  const float* __restrict__ as  = att_s + h * HID;
  const float* __restrict__ ad  = att_d + h * HID;
  float ss = 0.f, sd = 0.f;
  #pragma unroll
  for (int c = 0; c < HID; ++c) { float v = row[c]; ss += v * as[c]; sd += v * ad[c]; }
  a_s[idx] = ss; a_d[idx] = sd;
}

// ---------- layer-1 edge passes ----------
__global__ void k_max1(const long long* __restrict__ ei, const float* __restrict__ a_s,
                       const float* __restrict__ a_d, int* __restrict__ m1) {
  int e = blockIdx.x * blockDim.x + threadIdx.x;
  if (e >= ETOT) return;
  int s, d; edge_sd(ei, e, s, d);
  const float4* As = (const float4*)(a_s + (size_t)s * HEADS);
  const float4* Ad = (const float4*)(a_d + (size_t)d * HEADS);
  float4 s0 = As[0], s1 = As[1], d0 = Ad[0], d1 = Ad[1];
  float al[8] = { leaky(s0.x + d0.x), leaky(s0.y + d0.y), leaky(s0.z + d0.z), leaky(s0.w + d0.w),
                  leaky(s1.x + d1.x), leaky(s1.y + d1.y), leaky(s1.z + d1.z), leaky(s1.w + d1.w) };
  #pragma unroll
  for (int h = 0; h < HEADS; ++h) atomicMax(&m1[(size_t)d * HEADS + h], f2ord(al[h]));
}

__global__ void k_sum1(const long long* __restrict__ ei, const float* __restrict__ a_s,
                       const float* __restrict__ a_d, const int* __restrict__ m1,
                       float* __restrict__ s1) {
  int e = blockIdx.x * blockDim.x + threadIdx.x;
  if (e >= ETOT) return;
  int s, d; edge_sd(ei, e, s, d);
  const float4* As = (const float4*)(a_s + (size_t)s * HEADS);
  const float4* Ad = (const float4*)(a_d + (size_t)d * HEADS);
  float4 s0 = As[0], s1v = As[1], d0 = Ad[0], d1 = Ad[1];
  float al[8] = { leaky(s0.x + d0.x), leaky(s0.y + d0.y), leaky(s0.z + d0.z), leaky(s0.w + d0.w),
                  leaky(s1v.x + d1.x), leaky(s1v.y + d1.y), leaky(s1v.z + d1.z), leaky(s1v.w + d1.w) };
  #pragma unroll
  for (int h = 0; h < HEADS; ++h) {
    float ex = expf(al[h] - ord2f(m1[(size_t)d * HEADS + h]));
    atomicAdd(&s1[(size_t)d * HEADS + h], ex);
  }
}

// Weighted aggregation: one wave per edge, lane owns 8 channels of head lane>>2.
__global__ void k_agg1(const long long* __restrict__ ei, const float* __restrict__ a_s,
                       const float* __restrict__ a_d, const int* __restrict__ m1,
                       const float* __restrict__ s1, const float* __restrict__ h1,
                       float* __restrict__ agg1) {
  const int lane = threadIdx.x & 31;
  const int e    = (blockIdx.x * blockDim.x + threadIdx.x) >> 5;
  if (e >= ETOT) return;
  int s, d; edge_sd(ei, e, s, d);
  const int h = lane >> 2;
  float alpha = leaky(a_s[(size_t)s * HEADS + h] + a_d[(size_t)d * HEADS + h]);
  float w = expf(alpha - ord2f(m1[(size_t)d * HEADS + h])) / s1[(size_t)d * HEADS + h];
  const float4* hs = (const float4*)(h1 + (size_t)s * F1 + lane * 8);
  float4 v0 = hs[0], v1 = hs[1];
  float* dp = agg1 + (size_t)d * F1 + lane * 8;
  atomicAdd(dp + 0, v0.x * w); atomicAdd(dp + 1, v0.y * w);
  atomicAdd(dp + 2, v0.z * w); atomicAdd(dp + 3, v0.w * w);
  atomicAdd(dp + 4, v1.x * w); atomicAdd(dp + 5, v1.y * w);
  atomicAdd(dp + 6, v1.z * w); atomicAdd(dp + 7, v1.w * w);
}

// ---------- bias + ReLU + tiny GEMM2 + layer-2 attention logits (wave/node) ----------
__global__ void k_node2(const float* __restrict__ agg1, const float* __restrict__ b1,
                        const float* __restrict__ W2, const float* __restrict__ as2,
                        const float* __restrict__ ad2,
                        float* __restrict__ h2, float* __restrict__ a2s, float* __restrict__ a2d) {
  const int lane = threadIdx.x & 31;
  const int n    = (blockIdx.x * blockDim.x + threadIdx.x) >> 5;
  if (n >= N_NODES) return;
  const float4* rp = (const float4*)(agg1 + (size_t)n * F1 + lane * 8);
  const float4* bp = (const float4*)(b1 + lane * 8);
  float4 r0 = rp[0], r1 = rp[1], bb0 = bp[0], bb1 = bp[1];
  float vals[8] = { r0.x + bb0.x, r0.y + bb0.y, r0.z + bb0.z, r0.w + bb0.w,
                    r1.x + bb1.x, r1.y + bb1.y, r1.z + bb1.z, r1.w + bb1.w };
  float p0 = 0.f, p1 = 0.f;
  #pragma unroll
  for (int j = 0; j < 8; ++j) {
    float v = fmaxf(vals[j], 0.f);                 // ReLU
    int c = lane * 8 + j;
    p0 += v * W2[(size_t)c * CLASSES + 0];
    p1 += v * W2[(size_t)c * CLASSES + 1];
  }
  #pragma unroll
  for (int off = 16; off > 0; off >>= 1) {
    p0 += __shfl_xor(p0, off, 32);
    p1 += __shfl_xor(p1, off, 32);
  }
  if (lane == 0) {
    h2[(size_t)n * CLASSES + 0] = p0;
    h2[(size_t)n * CLASSES + 1] = p1;
    a2s[n] = p0 * as2[0] + p1 * as2[1];
    a2d[n] = p0 * ad2[0] + p1 * ad2[1];
  }
}

// ---------- layer-2 edge passes (1 head, 2 classes) ----------
__global__ void k_max2(const long long* __restrict__ ei, const float* __restrict__ a2s,
                       const float* __restrict__ a2d, int* __restrict__ m2) {
  int e = blockIdx.x * blockDim.x + threadIdx.x;
  if (e >= ETOT) return;
  int s, d; edge_sd(ei, e, s, d);
  atomicMax(&m2[d], f2ord(leaky(a2s[s] + a2d[d])));
}

__global__ void k_sum2(const long long* __restrict__ ei, const float* __restrict__ a2s,
                       const float* __restrict__ a2d, const int* __restrict__ m2,
                       float* __restrict__ s2) {
  int e = blockIdx.x * blockDim.x + threadIdx.x;
  if (e >= ETOT) return;
  int s, d; edge_sd(ei, e, s, d);
  atomicAdd(&s2[d], expf(leaky(a2s[s] + a2d[d]) - ord2f(m2[d])));
}

__global__ void k_agg2(const long long* __restrict__ ei, const float* __restrict__ a2s,
                       const float* __restrict__ a2d, const int* __restrict__ m2,
                       const float* __restrict__ s2, const float* __restrict__ h2,
                       float* __restrict__ out) {
  int e = blockIdx.x * blockDim.x + threadIdx.x;
  if (e >= ETOT) return;
  int s, d; edge_sd(ei, e, s, d);
  float w = expf(leaky(a2s[s] + a2d[d]) - ord2f(m2[d])) / s2[d];
  atomicAdd(&out[(size_t)d * CLASSES + 0], h2[(size_t)s * CLASSES + 0] * w);
  atomicAdd(&out[(size_t)d * CLASSES + 1], h2[(size_t)s * CLASSES + 1] * w);
}

// ---------- driver ----------
extern "C" void kernel_launch(void* const* d_in, const int* in_sizes, int n_in,
                              void* d_out, int out_size, void* d_ws, size_t ws_size,
                              hipStream_t stream) {
  const float*     x   = (const float*)d_in[0];
  const long long* ei  = (const long long*)d_in[1];   // int64 edge_index [2, 800000]
  const float*     W1  = (const float*)d_in[2];
  const float*     as1 = (const float*)d_in[3];
  const float*     ad1 = (const float*)d_in[4];
  const float*     b1  = (const float*)d_in[5];
  const float*     W2  = (const float*)d_in[6];
  const float*     as2 = (const float*)d_in[7];
  const float*     ad2 = (const float*)d_in[8];
  const float*     b2  = (const float*)d_in[9];
  float* out = (float*)d_out;

  char* ws = (char*)d_ws;
  size_t off = 0;
  auto alloc = [&](size_t bytes) -> void* {
    void* p = ws + off;
    off += (bytes + 255) & ~(size_t)255;
    return p;
  };
  float* h1   = (float*)alloc(sizeof(float) * (size_t)N_NODES * F1);
  float* agg1 = (float*)alloc(sizeof(float) * (size_t)N_NODES * F1);
  float* a_s1 = (float*)alloc(sizeof(float) * (size_t)N_NODES * HEADS);
  float* a_d1 = (float*)alloc(sizeof(float) * (size_t)N_NODES * HEADS);
  int*   m1   = (int*)  alloc(sizeof(int)   * (size_t)N_NODES * HEADS);
  float* s1   = (float*)alloc(sizeof(float) * (size_t)N_NODES * HEADS);
  float* h2   = (float*)alloc(sizeof(float) * (size_t)N_NODES * CLASSES);
  float* a2s  = (float*)alloc(sizeof(float) * (size_t)N_NODES);
  float* a2d  = (float*)alloc(sizeof(float) * (size_t)N_NODES);
  int*   m2   = (int*)  alloc(sizeof(int)   * (size_t)N_NODES);
  float* s2   = (float*)alloc(sizeof(float) * (size_t)N_NODES);
  (void)ws_size; (void)in_sizes; (void)n_in; (void)out_size;

  const int B = 256;
  // 0) init accumulators (covers largest buffer N*256)
  k_init<<<((size_t)N_NODES * F1 + B - 1) / B, B, 0, stream>>>(agg1, s1, m1, s2, m2, out, b2);
  // 1) WMMA GEMM: h1 = x @ W1   (3125*16 tiles, 8 waves/block)
  {
    int tiles = (N_NODES / 16) * (F1 / 16);
    k_gemm1<<<(tiles + 7) / 8, B, 0, stream>>>(x, W1, h1);
  }
  // 2) attention logits per (node, head)
  k_att1<<<(N_NODES * HEADS + B - 1) / B, B, 0, stream>>>(h1, as1, ad1, a_s1, a_d1);
  // 3-5) layer-1 segment softmax + aggregation
  k_max1<<<(ETOT + B - 1) / B, B, 0, stream>>>(ei, a_s1, a_d1, m1);
  k_sum1<<<(ETOT + B - 1) / B, B, 0, stream>>>(ei, a_s1, a_d1, m1, s1);
  k_agg1<<<((size_t)ETOT * 32 + B - 1) / B, B, 0, stream>>>(ei, a_s1, a_d1, m1, s1, h1, agg1);
  // 6) bias + ReLU + GEMM2 + layer-2 logits
  k_node2<<<((size_t)N_NODES * 32 + B - 1) / B, B, 0, stream>>>(agg1, b1, W2, as2, ad2, h2, a2s, a2d);
  // 7-9) layer-2 segment softmax + aggregation into d_out (pre-seeded with b2)
  k_max2<<<(ETOT + B - 1) / B, B, 0, stream>>>(ei, a2s, a2d, m2);
  k_sum2<<<(ETOT + B - 1) / B, B, 0, stream>>>(ei, a2s, a2d, m2, s2);
  k_agg2<<<(ETOT + B - 1) / B, B, 0, stream>>>(ei, a2s, a2d, m2, s2, h2, out);
}